// LSTM_23502061044161
// MI455X (gfx1250) — compile-verified
//
#include <hip/hip_runtime.h>
#include <stdint.h>

typedef uint32_t v4u __attribute__((ext_vector_type(4)));
typedef int32_t  v8i __attribute__((ext_vector_type(8)));
typedef int32_t  v4i __attribute__((ext_vector_type(4)));
typedef float    v2f __attribute__((ext_vector_type(2)));

#define LOG2E 1.4426950408889634f

#if __has_builtin(__builtin_amdgcn_tanhf)
__device__ __forceinline__ float ftanh(float x) { return __builtin_amdgcn_tanhf(x); }
__device__ __forceinline__ float fsigmoid(float x) {
  // sigmoid(x) = 0.5*tanh(x/2) + 0.5  (exact identity) -> one v_tanh_f32
  return fmaf(0.5f, __builtin_amdgcn_tanhf(0.5f * x), 0.5f);
}
#else
__device__ __forceinline__ float fsigmoid(float x) {
  return __builtin_amdgcn_rcpf(1.0f + __builtin_amdgcn_exp2f(-LOG2E * x));
}
__device__ __forceinline__ float ftanh(float x) {
  return fmaf(2.0f, fsigmoid(2.0f * x), -1.0f);
}
#endif

// Packed f32 FMA (VOP3P V_PK_FMA_F32), VGPR-pair operands only:
// gfx1250 rejects SGPR-pair sources with packed hi-half selects.
__device__ __forceinline__ v2f pk_fma(v2f a, v2f b, v2f c) {
  v2f d;
  asm("v_pk_fma_f32 %0, %1, %2, %3" : "=v"(d) : "v"(a), "v"(b), "v"(c));
  return d;
}

// Broadcast 32-bit value from lane L to all lanes via LDS-permute hardware
// (ds_swizzle BCASTX32: and=0x00, or=L, xor=0x00 -> offset = L<<5).
// No LDS memory touched; off the VALU issue port.
template <int L>
__device__ __forceinline__ float bcast(float v) {
  return __builtin_bit_cast(
      float, __builtin_amdgcn_ds_swizzle(__builtin_bit_cast(int, v), (L << 5)));
}

// Single-wave latency-optimized LSTM scan.
// Lane j (j<10) owns hidden unit j: i_j, f_j, g_j, o_j computed as four
// independent 6-deep v_pk_fma_f32 chains against h pairs broadcast to all
// lanes by ds_swizzle. Input sequence staged whole into LDS by one TDM op.
extern "C" __global__ __launch_bounds__(32, 1)
void lstm_scan_kernel(const float* __restrict__ x,     // [T]
                      const float* __restrict__ W_ih,  // [40] (IN==1)
                      const float* __restrict__ W_hh,  // [40,10] row-major
                      const float* __restrict__ b_ih,  // [40]
                      const float* __restrict__ b_hh,  // [40]
                      const float* __restrict__ W_lin, // [10]
                      const float* __restrict__ b_lin, // [1]
                      float* __restrict__ out,         // [1]
                      int T) {
  extern __shared__ float xs[];

  // ---- TDM: DMA the whole input sequence (T * 4 bytes) into LDS ----
  {
    uint64_t ga = (uint64_t)(uintptr_t)x;
    v4u g0;
    v8i g1;
    v4i g2 = {0, 0, 0, 0}, g3 = {0, 0, 0, 0};
    v8i g4 = {0, 0, 0, 0, 0, 0, 0, 0};
    uint32_t Tu = (uint32_t)T;
    // Group 0: count=1 | lds_addr=0 | global_addr | type=2 ("image")
    g0[0] = 1u;
    g0[1] = 0u;
    g0[2] = (uint32_t)(ga & 0xFFFFFFFFu);
    g0[3] = (uint32_t)((ga >> 32) & 0x01FFFFFFu) | (2u << 30);
    // Group 1: wg_mask=0, data_size=2 (4B), no flags/pad
    g1[0] = (int32_t)(2u << 16);
    g1[1] = (int32_t)((Tu & 0xFFFFu) << 16);                // tensor_dim0[15:0]
    g1[2] = (int32_t)(((Tu >> 16) & 0xFFFFu) | (1u << 16)); // dim0[31:16] | tensor_dim1=1
    g1[3] = (int32_t)((Tu & 0xFFFFu) << 16);                // tile_dim0 = T
    g1[4] = 1;                                              // tile_dim1 = 1, tile_dim2 = 0
    g1[5] = (int32_t)Tu;                                    // tensor_dim0_stride lo32
    g1[6] = 0;
    g1[7] = 0;
    __builtin_amdgcn_tensor_load_to_lds(g0, g1, g2, g3, g4, 0);
    __builtin_amdgcn_s_wait_tensorcnt(0);
  }

  const int tid = (int)threadIdx.x;
  const int j = tid < 10 ? tid : 9;  // lanes 10..31 duplicate lane 9 (no divergence, no OOB)
  const int ri = j, rf = j + 10, rg = j + 20, ro = j + 30;

  // Per-lane W_hh rows as packed pairs (VGPR pairs, live across the scan)
  v2f wi2[5], wf2[5], wg2[5], wo2[5];
#pragma unroll
  for (int p = 0; p < 5; ++p) {
    wi2[p].x = W_hh[ri * 10 + 2 * p];
    wi2[p].y = W_hh[ri * 10 + 2 * p + 1];
    wf2[p].x = W_hh[rf * 10 + 2 * p];
    wf2[p].y = W_hh[rf * 10 + 2 * p + 1];
    wg2[p].x = W_hh[rg * 10 + 2 * p];
    wg2[p].y = W_hh[rg * 10 + 2 * p + 1];
    wo2[p].x = W_hh[ro * 10 + 2 * p];
    wo2[p].y = W_hh[ro * 10 + 2 * p + 1];
  }
  // x-term pair {0, W_ih[row]} and bias pair {b_ih+b_hh, 0}: one packed FMA
  // with {xv,xv} yields {bias, wx*xv} as the chain seed.
  const v2f wxi2 = {0.0f, W_ih[ri]}, wxf2 = {0.0f, W_ih[rf]};
  const v2f wxg2 = {0.0f, W_ih[rg]}, wxo2 = {0.0f, W_ih[ro]};
  const v2f bi2 = {b_ih[ri] + b_hh[ri], 0.0f};
  const v2f bf2 = {b_ih[rf] + b_hh[rf], 0.0f};
  const v2f bg2 = {b_ih[rg] + b_hh[rg], 0.0f};
  const v2f bo2 = {b_ih[ro] + b_hh[ro], 0.0f};

  float c = 0.0f;
  v2f sh2[5];  // h pairs replicated in ALL lanes: {h[2p], h[2p+1]}
#pragma unroll
  for (int p = 0; p < 5; ++p) sh2[p] = (v2f){0.0f, 0.0f};

  auto step = [&](float xv) {
    v2f xv2 = {xv, xv};
    // Seed: {bias, wx*xv}; then 5 packed FMAs over h (4 independent chains)
    v2f ai = pk_fma(wxi2, xv2, bi2);
    v2f af = pk_fma(wxf2, xv2, bf2);
    v2f ag = pk_fma(wxg2, xv2, bg2);
    v2f ao = pk_fma(wxo2, xv2, bo2);
#pragma unroll
    for (int p = 0; p < 5; ++p) {
      ai = pk_fma(wi2[p], sh2[p], ai);
      af = pk_fma(wf2[p], sh2[p], af);
      ag = pk_fma(wg2[p], sh2[p], ag);
      ao = pk_fma(wo2[p], sh2[p], ao);
    }
    float gi = fsigmoid(ai.x + ai.y);
    float gf = fsigmoid(af.x + af.y);
    float gg = ftanh(ag.x + ag.y);
    float go = fsigmoid(ao.x + ao.y);
    c = fmaf(gf, c, gi * gg);
    float h = go * ftanh(c);
    // Broadcast h_k (owner lane k) to all lanes via LDS-permute; results land
    // directly in the VGPR pair halves the next step's packed FMAs read.
    sh2[0].x = bcast<0>(h);
    sh2[0].y = bcast<1>(h);
    sh2[1].x = bcast<2>(h);
    sh2[1].y = bcast<3>(h);
    sh2[2].x = bcast<4>(h);
    sh2[2].y = bcast<5>(h);
    sh2[3].x = bcast<6>(h);
    sh2[3].y = bcast<7>(h);
    sh2[4].x = bcast<8>(h);
    sh2[4].y = bcast<9>(h);
  };

  // Hot loop: x from LDS, one float4 prefetched a block ahead (ds latency hidden)
  const float4* x4 = reinterpret_cast<const float4*>(xs);
  const int nb = T >> 2;
  if (nb > 0) {
    float4 cur = x4[0];
    for (int b = 0; b < nb; ++b) {
      float4 nxt = x4[(b + 1 < nb) ? (b + 1) : b];
      step(cur.x);
      step(cur.y);
      step(cur.z);
      step(cur.w);
      cur = nxt;
    }
  }
  for (int t = nb << 2; t < T; ++t) step(xs[t]);

  // Linear head on final h (sh2 pairs are replicated in every lane)
  if (tid == 0) {
    float acc = b_lin[0];
#pragma unroll
    for (int p = 0; p < 5; ++p) {
      acc = fmaf(W_lin[2 * p], sh2[p].x, acc);
      acc = fmaf(W_lin[2 * p + 1], sh2[p].y, acc);
    }
    out[0] = acc;
  }
}

extern "C" void kernel_launch(void* const* d_in, const int* in_sizes, int n_in,
                              void* d_out, int out_size, void* d_ws, size_t ws_size,
                              hipStream_t stream) {
  const float* x    = (const float*)d_in[0];
  const float* Wih  = (const float*)d_in[1];
  const float* Whh  = (const float*)d_in[2];
  const float* bih  = (const float*)d_in[3];
  const float* bhh  = (const float*)d_in[4];
  const float* Wlin = (const float*)d_in[5];
  const float* blin = (const float*)d_in[6];
  const int T = in_sizes[0];  // IN == 1, so input_seq has T elements

  const size_t shmem = (size_t)T * sizeof(float);  // 128 KB for T=32768; LDS is 320 KB/WGP
  (void)hipFuncSetAttribute((const void*)lstm_scan_kernel,
                            hipFuncAttributeMaxDynamicSharedMemorySize,
                            (int)shmem);  // best-effort, idempotent

  lstm_scan_kernel<<<1, 32, shmem, stream>>>(x, Wih, Whh, bih, bhh, Wlin, blin,
                                             (float*)d_out, T);
}